// VocabularyProjectionFFN_10969346474687
// MI455X (gfx1250) — compile-verified
//
#include <hip/hip_runtime.h>
#include <hip/hip_bf16.h>

typedef __attribute__((ext_vector_type(2))) float v2f;
typedef __attribute__((ext_vector_type(8))) float v8f;

#define V_    32000
#define NH_   32
#define HD_   32
#define C_    1024
#define BT_   128
#define R_    4096          // BT_*NH_ rows
#define SPLIT 8             // waves per workgroup, split over vocab
#define VW    (V_ / SPLIT)  // 4000 vocab entries per wave

// ---------------------------------------------------------------------------
// Kernel 1: ffn_scaled[r][d] = (sum_h x[r*32+h]*W[d*32+h] + b[d]) / clip(temp[r%32],0.1)
// ---------------------------------------------------------------------------
__global__ __launch_bounds__(256) void k_ffn(const float* __restrict__ x,
                                             const float* __restrict__ W,
                                             const float* __restrict__ b,
                                             const float* __restrict__ temps,
                                             float* __restrict__ ffn) {
  int tid = blockIdx.x * blockDim.x + threadIdx.x;   // r*32 + d
  int r = tid >> 5, d = tid & 31;
  const float4* x4 = (const float4*)x + r * 8;
  const float4* W4 = (const float4*)W + d * 8;
  float s = 0.f;
#pragma unroll
  for (int i = 0; i < 8; ++i) {
    float4 a = x4[i], w = W4[i];
    s = fmaf(a.x, w.x, fmaf(a.y, w.y, fmaf(a.z, w.z, fmaf(a.w, w.w, s))));
  }
  s += b[d];
  float t = fmaxf(temps[r & 31], 0.1f);
  ffn[tid] = s / t;
}

// ---------------------------------------------------------------------------
// Kernel 2: vs[h][v] = sum_d vocab_emb[v*1024 + h*32 + d]   (131 MB streamed once)
// ---------------------------------------------------------------------------
__global__ __launch_bounds__(256) void k_vs(const float* __restrict__ ve,
                                            float* __restrict__ vs) {
  int tid = blockIdx.x * blockDim.x + threadIdx.x;   // v*32 + h
  int v = tid >> 5, h = tid & 31;
  const float4* p = (const float4*)ve + v * 256 + h * 8;
  float s = 0.f;
#pragma unroll
  for (int i = 0; i < 8; ++i) { float4 a = p[i]; s += (a.x + a.y) + (a.z + a.w); }
  vs[h * V_ + v] = s;
}

// ---------------------------------------------------------------------------
// Kernel 3: fused logits -> online softmax -> weighted sum.
// 16 rows per workgroup; SPLIT waves split the vocab axis (flash-decoding),
// partial (m, l, acc) merged through LDS at the end.
// WMMA f32 16x16x4; layouts per CDNA5 ISA 7.12.2:
//   A  (16x4):  elem(Vj, lane) = (M=lane%16, K=j+2*(lane>=16))
//   B  (4x16):  elem(Vj, lane) = (K=j+2*(lane>=16), N=lane%16)
//   C/D(16x16): elem(Vg, lane) = (M=g+8*(lane>=16), N=lane%16)
// ---------------------------------------------------------------------------
__global__ __launch_bounds__(32 * SPLIT) void k_main(const float* __restrict__ ffn,
                                                     const float* __restrict__ vs,
                                                     float* __restrict__ out) {
  __shared__ float lds_p[SPLIT][16 * 17];  // per-wave P tile [v_local][r], stride 17
  __shared__ float lds_s[SPLIT][16];       // per-wave rescale exchange
  __shared__ float lds_m[SPLIT][16];       // merge: per-wave row max
  __shared__ float lds_l[SPLIT][16];       // merge: per-wave row denom
  __shared__ float lds_o[16 * 33];         // merged out tile [r][h], stride 33

  const int lane = threadIdx.x & 31;
  const int wv   = threadIdx.x >> 5;       // 0..SPLIT-1
  const int col  = lane & 15;              // N index (row r for C/D, v for B loads)
  const int half = lane >> 4;
  const int r0   = blockIdx.x * 16;

  // zero the merged-output tile (atomically accumulated later)
  for (int i = threadIdx.x; i < 16 * 33; i += 32 * SPLIT) lds_o[i] = 0.f;

  // A tiles (logits): rows = ffn_scaled[r0..r0+15], K = h (32) -> 8 chained WMMAs
  v2f A[8];
#pragma unroll
  for (int kk = 0; kk < 8; ++kk) {
    const float* pa = ffn + (r0 + col) * HD_ + 4 * kk + 2 * half;
    A[kk].x = pa[0]; A[kk].y = pa[1];
  }

  v8f acc0 = {}, acc1 = {};                // out^T accumulators, h-tiles 0 / 1
  float mh[8], lh[8];                      // online-softmax state, row = g + 8*half
#pragma unroll
  for (int g = 0; g < 8; ++g) { mh[g] = -__builtin_inff(); lh[g] = 0.f; }

  const int vbeg = wv * VW;
  for (int vt = 0; vt < VW / 16; ++vt) {
    const int vb = vbeg + vt * 16;
    // ---- logits tile S[r][v] = (ffn/temp) . vocab_sum, K = 32 ----
    v8f S = {};
#pragma unroll
    for (int kk = 0; kk < 8; ++kk) {
      v2f Bt;
      const float* pb = vs + (4 * kk + 2 * half) * V_ + vb + col;
      Bt.x = pb[0];      // K = 4kk + 2*half
      Bt.y = pb[V_];     // K = 4kk + 2*half + 1
      S = __builtin_amdgcn_wmma_f32_16x16x4_f32(false, A[kk], false, Bt,
                                                (short)0, S, false, false);
    }

    // ---- online softmax (per row = g + 8*half, uniform within half-wave) ----
    float p[8], scale_h[8];
#pragma unroll
    for (int g = 0; g < 8; ++g) {
      float tm = S[g];
#pragma unroll
      for (int m = 1; m < 16; m <<= 1) tm = fmaxf(tm, __shfl_xor(tm, m, 16));
      float mn = fmaxf(mh[g], tm);
      float sc = __expf(mh[g] - mn);       // 0 on first tile (-inf)
      float pe = __expf(S[g] - mn);
      float ts = pe;
#pragma unroll
      for (int m = 1; m < 16; m <<= 1) ts += __shfl_xor(ts, m, 16);
      lh[g] = lh[g] * sc + ts;
      mh[g] = mn;
      p[g] = pe;
      scale_h[g] = sc;
    }

    // ---- stage rescale factors + P tile through LDS (per-wave region) ----
    if (lane == 0) {
#pragma unroll
      for (int g = 0; g < 8; ++g) lds_s[wv][g] = scale_h[g];       // rows 0..7
    }
    if (lane == 16) {
#pragma unroll
      for (int g = 0; g < 8; ++g) lds_s[wv][8 + g] = scale_h[g];   // rows 8..15
    }
#pragma unroll
    for (int g = 0; g < 8; ++g)
      lds_p[wv][col * 17 + g + 8 * half] = p[g];  // [v_local=col][r=g+8*half]
    __syncthreads();

    float sc = lds_s[wv][col];             // rescale for row N = col
#pragma unroll
    for (int g = 0; g < 8; ++g) { acc0[g] *= sc; acc1[g] *= sc; }

    // ---- accumulate out^T[h][r] += vs[h][v] * P[v][r], K = 16 ----
#pragma unroll
    for (int kk = 0; kk < 4; ++kk) {
      int k0 = 4 * kk + 2 * half;
      v2f Bt;                              // B = P^T from LDS
      Bt.x = lds_p[wv][k0 * 17 + col];
      Bt.y = lds_p[wv][(k0 + 1) * 17 + col];
      v2f Aa, Ab;                          // A = vocab_sum rows (L2-resident)
      const float* pa0 = vs + col * V_ + vb + k0;          // h = col      (tile 0)
      const float* pa1 = vs + (16 + col) * V_ + vb + k0;   // h = 16+col   (tile 1)
      Aa.x = pa0[0]; Aa.y = pa0[1];
      Ab.x = pa1[0]; Ab.y = pa1[1];
      acc0 = __builtin_amdgcn_wmma_f32_16x16x4_f32(false, Aa, false, Bt,
                                                   (short)0, acc0, false, false);
      acc1 = __builtin_amdgcn_wmma_f32_16x16x4_f32(false, Ab, false, Bt,
                                                   (short)0, acc1, false, false);
    }
    __syncthreads();                       // WAR on lds_p / lds_s
  }

  // ---- merge the SPLIT partial softmax states ----
  if (lane == 0) {
#pragma unroll
    for (int g = 0; g < 8; ++g) { lds_m[wv][g] = mh[g]; lds_l[wv][g] = lh[g]; }
  }
  if (lane == 16) {
#pragma unroll
    for (int g = 0; g < 8; ++g) { lds_m[wv][8 + g] = mh[g]; lds_l[wv][8 + g] = lh[g]; }
  }
  __syncthreads();

  float M = -__builtin_inff();
#pragma unroll
  for (int w = 0; w < SPLIT; ++w) M = fmaxf(M, lds_m[w][col]);
  float L = 0.f;
#pragma unroll
  for (int w = 0; w < SPLIT; ++w) L += lds_l[w][col] * __expf(lds_m[w][col] - M);
  float f = __expf(lds_m[wv][col] - M) / L;   // this wave's contribution factor

#pragma unroll
  for (int g = 0; g < 8; ++g) {
    int h0 = g + 8 * half;                    // M index of D
    atomicAdd(&lds_o[col * 33 + h0],      acc0[g] * f);   // ds_add_f32
    atomicAdd(&lds_o[col * 33 + 16 + h0], acc1[g] * f);
  }
  __syncthreads();

  // coalesced store: 512 outputs, 256 threads x 2
  int i0 = threadIdx.x;
  out[r0 * 32 + i0]       = lds_o[(i0 >> 5) * 33 + (i0 & 31)];
  int i1 = i0 + 256;
  out[r0 * 32 + i1]       = lds_o[(i1 >> 5) * 33 + (i1 & 31)];
}

// ---------------------------------------------------------------------------
extern "C" void kernel_launch(void* const* d_in, const int* in_sizes, int n_in,
                              void* d_out, int out_size, void* d_ws, size_t ws_size,
                              hipStream_t stream) {
  const float* x     = (const float*)d_in[0];   // (B,T,C)   131072
  const float* W     = (const float*)d_in[1];   // (32,32)   1024
  const float* b     = (const float*)d_in[2];   // (32,)
  const float* temps = (const float*)d_in[3];   // (32,)
  const float* ve    = (const float*)d_in[4];   // (32000,1024)
  float* out = (float*)d_out;

  float* vs  = (float*)d_ws;                    // 32*32000 floats = 4 MB
  float* ffn = vs + (size_t)NH_ * V_;           // 4096*32 floats

  k_vs  <<<(V_ * 32) / 256, 256, 0, stream>>>(ve, vs);
  k_ffn <<<(R_ * 32) / 256, 256, 0, stream>>>(x, W, b, temps, ffn);
  k_main<<<R_ / 16, 32 * SPLIT, 0, stream>>>(ffn, vs, out);
}